// RTDETRV2MultiScaleDeformableAttention_43748536877147
// MI455X (gfx1250) — compile-verified
//
#include <hip/hip_runtime.h>
#include <math.h>

// ---------------------------------------------------------------------------
// RT-DETRv2 multi-scale deformable attention, MI455X (gfx1250, wave32, WMMA).
//
// Roofline: the value projection GEMM (134400x256x256 fp32, 17.6 GFLOP) moves
// ~280 MB HBM => ~12us at 23.3 TB/s; intensity ~63 FLOP/B => bandwidth bound.
// Inputs are fp32, so V_WMMA_F32_16X16X4_F32 is the right instruction: full
// precision, no conversion pass, and f32 WMMA rate far exceeds the ~1.5 TFLOP/s
// the bandwidth roof requires. A-panels are staged in LDS (one 16x256 panel per
// block, shared by all 8 waves) so each element of the 137 MB activation matrix
// is read from global exactly once; fragments come back via ds_load_2addr_b64
// with a padded stride that hits all 64 LDS banks exactly once.
// ---------------------------------------------------------------------------

typedef __attribute__((ext_vector_type(2))) float v2f;
typedef __attribute__((ext_vector_type(8))) float v8f;

#define BATCH     16
#define LEN_Q     300
#define D_MODEL   256
#define N_HEADS   8
#define HEAD_DIM  32
#define N_LEVELS  3
#define N_POINTS  4
#define N_LP      12
#define N_TOTAL   8400   // 80*80 + 40*40 + 20*20
#define KD        256    // inner dim of every GEMM in this model
#define LDS_STRIDE (KD + 4)  // pad so b64 fragment reads are bank-conflict-free

// ---------------------------------------------------------------------------
// fp32 WMMA GEMM with fused bias:  C[M,N] = A[M,KD] @ B[KD,N] + bias[N]
// Row-major. M%16==0, N%32==0. One block = one 16-row M tile; wave w owns the
// 32-wide N tile w (ntiles <= 8 in all uses). A panel staged via LDS.
//
// V_WMMA_F32_16X16X4_F32 operand layout (ISA 7.12.2, 32-bit A 16x4):
//   lane = 16*hi + lo (lo=0..15, hi=0..1)
//   A frag component j  : A[m = lo][k + 2*hi + j]           (j = 0,1)
//   B frag component j  : B[k + 2*hi + j][n = ntile + lo]
//   C VGPR r            : C[m = r + 8*hi][n = ntile + lo]
// ---------------------------------------------------------------------------
__global__ __launch_bounds__(256) void gemm_bias_wmma_f32(
    const float* __restrict__ A, const float* __restrict__ Bm,
    const float* __restrict__ bias, float* __restrict__ C,
    int M, int N)
{
    __shared__ float As[16 * LDS_STRIDE];   // 16.6 KB

    const int tid  = threadIdx.x;
    const int lane = tid & 31;
    const int wave = tid >> 5;
    const int lo   = lane & 15;
    const int hi   = lane >> 4;
    const int mt   = blockIdx.x;

    // Cooperative A-panel stage: 16 rows x 256 cols, thread t copies 16 floats
    // (4x float4). Row stride padded to 260 floats; (r*260 + c)*4 stays 16B
    // aligned because c is a multiple of 16 and 260*4 = 65*16.
    {
        const int r = tid >> 4;            // 0..15
        const int c = (tid & 15) << 4;     // 0,16,...,240
        const float4* src = (const float4*)(A + (size_t)(mt * 16 + r) * KD + c);
        float4* dst = (float4*)&As[r * LDS_STRIDE + c];
#pragma unroll
        for (int i = 0; i < 4; ++i) dst[i] = src[i];
    }
    __syncthreads();

    const int ntiles = N >> 5;             // 32-wide wave tiles along N
    const int nt = wave;
    if (nt < ntiles) {
        const int    n0   = nt * 32 + lo;
        const float* Arow = &As[lo * LDS_STRIDE + 2 * hi];
        // Two live 32-bit offsets so all four B loads stay in the
        // saddr + scaled-voffset (+imm) form: row k+2hi and row k+2hi+1.
        int bo0 = (2 * hi) * N + n0;
        int bo1 = bo0 + N;

        v8f c0 = {};
        v8f c1 = {};
#pragma unroll 4
        for (int k = 0; k < KD; k += 4) {
            // A fragment from LDS: ds_load_b64, all banks distinct.
            v2f a = *(const v2f*)(Arow + k);

            v2f b0, b1;
            b0.x = Bm[bo0];
            b0.y = Bm[bo1];
            b1.x = Bm[bo0 + 16];
            b1.y = Bm[bo1 + 16];
            bo0 += 4 * N;
            bo1 += 4 * N;

            // (neg_a, A, neg_b, B, c_mod, C, reuse_a, reuse_b)
            c0 = __builtin_amdgcn_wmma_f32_16x16x4_f32(false, a, false, b0,
                                                       (short)0, c0, false, false);
            c1 = __builtin_amdgcn_wmma_f32_16x16x4_f32(false, a, false, b1,
                                                       (short)0, c1, false, false);
        }

        const float bb0 = bias[n0];
        const float bb1 = bias[n0 + 16];
        int co = (mt * 16 + 8 * hi) * N + n0;   // < 2^31 for all uses here
#pragma unroll
        for (int r = 0; r < 8; ++r) {
            C[co + r * N]      = c0[r] + bb0;
            C[co + r * N + 16] = c1[r] + bb1;
        }
    }
}

// ---------------------------------------------------------------------------
// Deformable sampling core. One wave per (b, q, head); lane = channel d
// (HEAD_DIM == 32 == wave32). Softmax over the 12 attention logits is
// recomputed per lane (wave-uniform, negligible). Each corner gather reads 32
// consecutive floats (value kept in (B, N_TOTAL, 256) layout) => coalesced.
// ---------------------------------------------------------------------------
__global__ __launch_bounds__(256) void deform_sample_core(
    const float* __restrict__ value,   // (B, N_TOTAL, 256)
    const float* __restrict__ refp,    // (B, LEN_Q, 3, 2)
    const float* __restrict__ off,     // (B, LEN_Q, 192)  = (H,3,4,2)
    const float* __restrict__ logits,  // (B, LEN_Q, 96)   = (H,12)
    float* __restrict__ core)          // (B, LEN_Q, 256)
{
    const int lane = threadIdx.x & 31;
    const int wave = threadIdx.x >> 5;
    const int gid  = blockIdx.x * 8 + wave;
    if (gid >= BATCH * LEN_Q * N_HEADS) return;

    const int h = gid % N_HEADS;
    const int q = (gid / N_HEADS) % LEN_Q;
    const int b = gid / (N_HEADS * LEN_Q);
    const int row = b * LEN_Q + q;

    // Softmax over this head's 12 logits.
    const float* lg = logits + (size_t)row * (N_HEADS * N_LP) + h * N_LP;
    float w[N_LP];
    float mx = lg[0];
#pragma unroll
    for (int i = 1; i < N_LP; ++i) mx = fmaxf(mx, lg[i]);
    float s = 0.f;
#pragma unroll
    for (int i = 0; i < N_LP; ++i) { w[i] = __expf(lg[i] - mx); s += w[i]; }
    const float inv = 1.0f / s;

    const int Hs[N_LEVELS]     = {80, 40, 20};
    const int Ws[N_LEVELS]     = {80, 40, 20};
    const int starts[N_LEVELS] = {0, 6400, 8000};

    const float* vbase = value + (size_t)b * N_TOTAL * D_MODEL + h * HEAD_DIM + lane;
    const float* rp    = refp + (size_t)row * (N_LEVELS * 2);
    const float* op    = off + (size_t)row * (N_HEADS * N_LP * 2) + h * (N_LP * 2);

    float acc = 0.f;
#pragma unroll
    for (int lid = 0; lid < N_LEVELS; ++lid) {
        const int Hh = Hs[lid], Ww = Ws[lid], st = starts[lid];
        const float rx = rp[lid * 2 + 0];
        const float ry = rp[lid * 2 + 1];
        const float invW = 1.0f / (float)Ww;
        const float invH = 1.0f / (float)Hh;
#pragma unroll
        for (int p = 0; p < N_POINTS; ++p) {
            const float ox = op[(lid * N_POINTS + p) * 2 + 0];
            const float oy = op[(lid * N_POINTS + p) * 2 + 1];
            // loc = ref + off/[W,H];  ix = loc_x*W - 0.5  (== ((2loc-1+1)W-1)/2)
            const float ix = (rx + ox * invW) * (float)Ww - 0.5f;
            const float iy = (ry + oy * invH) * (float)Hh - 0.5f;
            const float fx0 = floorf(ix), fy0 = floorf(iy);
            const int ix0 = (int)fx0, iy0 = (int)fy0;
            const int ix1 = ix0 + 1,  iy1 = iy0 + 1;
            const float fx = ix - fx0, fy = iy - fy0;

            const bool x0ok = (ix0 >= 0) & (ix0 < Ww);
            const bool x1ok = (ix1 >= 0) & (ix1 < Ww);
            const bool y0ok = (iy0 >= 0) & (iy0 < Hh);
            const bool y1ok = (iy1 >= 0) & (iy1 < Hh);
            const int cx0 = min(max(ix0, 0), Ww - 1);
            const int cx1 = min(max(ix1, 0), Ww - 1);
            const int cy0 = min(max(iy0, 0), Hh - 1);
            const int cy1 = min(max(iy1, 0), Hh - 1);

            float v00 = 0.f, v01 = 0.f, v10 = 0.f, v11 = 0.f;
            if (y0ok & x0ok) v00 = vbase[(size_t)(st + cy0 * Ww + cx0) * D_MODEL];
            if (y0ok & x1ok) v01 = vbase[(size_t)(st + cy0 * Ww + cx1) * D_MODEL];
            if (y1ok & x0ok) v10 = vbase[(size_t)(st + cy1 * Ww + cx0) * D_MODEL];
            if (y1ok & x1ok) v11 = vbase[(size_t)(st + cy1 * Ww + cx1) * D_MODEL];

            const float sam = (1.f - fx) * (1.f - fy) * v00 + fx * (1.f - fy) * v01
                            + (1.f - fx) * fy * v10 + fx * fy * v11;
            acc = fmaf(w[lid * N_POINTS + p] * inv, sam, acc);
        }
    }
    core[(size_t)row * D_MODEL + h * HEAD_DIM + lane] = acc;
}

// ---------------------------------------------------------------------------
// Launch: 4 WMMA GEMMs + 1 sampling kernel, stream-ordered dependencies.
// Workspace: value 137.6MB | off 3.7MB | attn 1.8MB | core 4.9MB  (~148MB)
// ---------------------------------------------------------------------------
extern "C" void kernel_launch(void* const* d_in, const int* in_sizes, int n_in,
                              void* d_out, int out_size, void* d_ws, size_t ws_size,
                              hipStream_t stream) {
    const float* query  = (const float*)d_in[0];
    const float* refp   = (const float*)d_in[1];
    const float* inflat = (const float*)d_in[2];
    const float* W_off  = (const float*)d_in[3];
    const float* b_off  = (const float*)d_in[4];
    const float* W_attn = (const float*)d_in[5];
    const float* b_attn = (const float*)d_in[6];
    const float* W_val  = (const float*)d_in[7];
    const float* b_val  = (const float*)d_in[8];
    const float* W_out  = (const float*)d_in[9];
    const float* b_out  = (const float*)d_in[10];
    float* out = (float*)d_out;

    float* value = (float*)d_ws;                                     // B*8400*256
    float* off   = value + (size_t)BATCH * N_TOTAL * D_MODEL;        // B*300*192
    float* attn  = off   + (size_t)BATCH * LEN_Q * (N_HEADS*N_LP*2); // B*300*96
    float* core  = attn  + (size_t)BATCH * LEN_Q * (N_HEADS*N_LP);   // B*300*256

    // 1) value = input_flatten @ W_val + b_val : M=134400 N=256 (8 ntiles/block)
    gemm_bias_wmma_f32<<<dim3(8400, 1), 256, 0, stream>>>(
        inflat, W_val, b_val, value, BATCH * N_TOTAL, 256);
    // 2) off = query @ W_off + b_off : M=4800 N=192 (6 active waves/block)
    gemm_bias_wmma_f32<<<dim3(300, 1), 256, 0, stream>>>(
        query, W_off, b_off, off, BATCH * LEN_Q, 192);
    // 3) attn logits = query @ W_attn + b_attn : M=4800 N=96 (3 active waves)
    gemm_bias_wmma_f32<<<dim3(300, 1), 256, 0, stream>>>(
        query, W_attn, b_attn, attn, BATCH * LEN_Q, 96);
    // 4) deformable sampling + per-head softmax : 16*300*8 = 38400 waves
    deform_sample_core<<<4800, 256, 0, stream>>>(value, refp, off, attn, core);
    // 5) out = core @ W_out + b_out : M=4800 N=256
    gemm_bias_wmma_f32<<<dim3(300, 1), 256, 0, stream>>>(
        core, W_out, b_out, out, BATCH * LEN_Q, 256);
}